// MemAELoss_54013508714894
// MI455X (gfx1250) — compile-verified
//
#include <hip/hip_runtime.h>
#include <math.h>

// ---------------- problem geometry ----------------
#define MSE_N      6291456      // 32*3*256*256
#define ATT_ROWS   8192
#define ATT_COLS   2000
#define N_MEM      2000
#define D_MEM      256
#define TILE       16
#define NTILE      125          // 2000 / 16 (exact)
#define NTILE_PAIRS 7875        // 125*126/2 upper-triangle tile pairs (incl. diagonal tiles)
#define MSE_BLOCKS 1024
#define REG_PARAM  0.0002f
#define COS_EPS    1e-8f

// ---------------- workspace layout (float offsets) ----------------
#define MSE_OFF 0                       // 1024 partials
#define REG_OFF 1024                    // 8192 partials
#define COS_OFF (1024 + 8192)           // 7875 partials
#define INV_OFF 17152                   // 2000 inverse norms (64-aligned)

typedef float v2f __attribute__((ext_vector_type(2)));
typedef float v8f __attribute__((ext_vector_type(8)));

// ---------------------------------------------------------------------------
// Kernel 1: MSE partial sums. Grid-stride over float4, block tree-reduce.
// ---------------------------------------------------------------------------
__global__ void mse_kernel(const float* __restrict__ out,
                           const float* __restrict__ gt,
                           float* __restrict__ partial) {
    __shared__ float red[256];
    const int tid = threadIdx.x;
    const size_t stride = (size_t)gridDim.x * blockDim.x;
    size_t i = (size_t)blockIdx.x * blockDim.x + tid;

    const float4* o4 = (const float4*)out;
    const float4* g4 = (const float4*)gt;
    const size_t n4 = MSE_N / 4;

    float acc = 0.0f;
    for (; i < n4; i += stride) {
        float4 a = o4[i];
        float4 b = g4[i];
        float dx = b.x - a.x, dy = b.y - a.y, dz = b.z - a.z, dw = b.w - a.w;
        acc += dx * dx + dy * dy + dz * dz + dw * dw;
    }
    red[tid] = acc;
    __syncthreads();
    for (int s = 128; s > 0; s >>= 1) {
        if (tid < s) red[tid] += red[tid + s];
        __syncthreads();
    }
    if (tid == 0) partial[blockIdx.x] = red[0];
}

// ---------------------------------------------------------------------------
// Kernel 2: entropy regularizer, one 256-thread block per row of att.
// row value = S/Z - log(Z)  with t=x-m, Z=sum(e^t), S=sum(t e^t).
// Row cached in registers (8 elements/thread); single pass over HBM.
// ---------------------------------------------------------------------------
__global__ void entropy_kernel(const float* __restrict__ att,
                               float* __restrict__ partial) {
    __shared__ float red[256];
    const int tid = threadIdx.x;
    const float* x = att + (size_t)blockIdx.x * ATT_COLS;

    float v[8];
    float m = -3.402823466e38f;
#pragma unroll
    for (int k = 0; k < 8; ++k) {
        int j = tid + k * 256;
        v[k] = (j < ATT_COLS) ? x[j] : -3.402823466e38f;
        m = fmaxf(m, v[k]);
    }
    red[tid] = m;
    __syncthreads();
    for (int s = 128; s > 0; s >>= 1) {
        if (tid < s) red[tid] = fmaxf(red[tid], red[tid + s]);
        __syncthreads();
    }
    m = red[0];
    __syncthreads();

    float Z = 0.0f, S = 0.0f;
#pragma unroll
    for (int k = 0; k < 8; ++k) {
        int j = tid + k * 256;
        if (j < ATT_COLS) {
            float t = v[k] - m;
            float e = expf(t);
            Z += e;
            S += t * e;
        }
    }
    red[tid] = Z;
    __syncthreads();
    for (int s = 128; s > 0; s >>= 1) {
        if (tid < s) red[tid] += red[tid + s];
        __syncthreads();
    }
    Z = red[0];
    __syncthreads();
    red[tid] = S;
    __syncthreads();
    for (int s = 128; s > 0; s >>= 1) {
        if (tid < s) red[tid] += red[tid + s];
        __syncthreads();
    }
    if (tid == 0) partial[blockIdx.x] = red[0] / Z - logf(Z);
}

// ---------------------------------------------------------------------------
// Kernel 3: inverse row norms of mem. One 256-thread block per row (D=256).
// ---------------------------------------------------------------------------
__global__ void invnorm_kernel(const float* __restrict__ mem,
                               float* __restrict__ inv) {
    __shared__ float red[256];
    const int tid = threadIdx.x;
    float v = mem[(size_t)blockIdx.x * D_MEM + tid];
    red[tid] = v * v;
    __syncthreads();
    for (int s = 128; s > 0; s >>= 1) {
        if (tid < s) red[tid] += red[tid + s];
        __syncthreads();
    }
    if (tid == 0) {
        float nrm = fmaxf(sqrtf(red[0]), COS_EPS);
        inv[blockIdx.x] = 1.0f / nrm;
    }
}

// ---------------------------------------------------------------------------
// Kernel 4: Gram triu-sum via v_wmma_f32_16x16x4_f32.
// One wave (32 threads) per 16x16 tile pair (I<=J), K accumulated 4 at a time.
// A-tile and B-tile lane layouts coincide because Gram = U*U^T:
//   lane l (hi = l>>4, m = l&15) loads row (tile*16 + m), cols k+2*hi, k+2*hi+1.
// C/D layout: VGPR v, lane l -> element (row = v + 8*hi, col = l&15).
// Strict-upper mask: diagonal tiles keep (row < col) only; I<J tiles keep all.
// Inverse norms are applied at drain time (raw dot * inv_i * inv_j).
// ---------------------------------------------------------------------------
__global__ void gram_kernel(const float* __restrict__ mem,
                            const float* __restrict__ inv,
                            float* __restrict__ partial) {
    const int lane = threadIdx.x;   // 0..31
    const int t = blockIdx.x;

    // Map linear tile-pair index -> (I, J) with I <= J (scalar, uniform).
    int rem = t, I = 0;
    for (int i = 0; i < NTILE; ++i) {
        int len = NTILE - i;
        if (rem < len) { I = i; break; }
        rem -= len;
    }
    const int J = I + rem;

    const int m  = lane & 15;
    const int hi = lane >> 4;

    const float* arow = mem + (size_t)(I * TILE + m) * D_MEM + 2 * hi;
    const float* brow = mem + (size_t)(J * TILE + m) * D_MEM + 2 * hi;

    v8f acc = {0.f, 0.f, 0.f, 0.f, 0.f, 0.f, 0.f, 0.f};
#pragma unroll 4
    for (int k = 0; k < D_MEM; k += 4) {
        v2f a = *(const v2f*)(arow + k);   // 8-byte aligned global_load_b64
        v2f b = *(const v2f*)(brow + k);
        acc = __builtin_amdgcn_wmma_f32_16x16x4_f32(
            /*neg_a=*/false, a, /*neg_b=*/false, b,
            /*c_mod=*/(short)0, acc, /*reuse_a=*/false, /*reuse_b=*/false);
    }

    // Drain with triu mask + normalization.
    const int n = lane & 15;
    const float invJn = inv[J * TILE + n];
    float local = 0.0f;
#pragma unroll
    for (int v = 0; v < 8; ++v) {
        int mrow = v + 8 * hi;
        bool keep = (I < J) || (mrow < n);
        if (keep) local += acc[v] * inv[I * TILE + mrow] * invJn;
    }

    __shared__ float red[32];
    red[lane] = local;
    __syncthreads();
    for (int s = 16; s > 0; s >>= 1) {
        if (lane < s) red[lane] += red[lane + s];
        __syncthreads();
    }
    if (lane == 0) partial[t] = red[0];
}

// ---------------------------------------------------------------------------
// Kernel 5: deterministic final reduction of all partials + combine.
// ---------------------------------------------------------------------------
__global__ void final_kernel(const float* __restrict__ ws,
                             float* __restrict__ out) {
    __shared__ float red[256];
    const int tid = threadIdx.x;
    float mse = 0.0f, reg = 0.0f, cosv = 0.0f;

    float s = 0.0f;
    for (int i = tid; i < MSE_BLOCKS; i += 256) s += ws[MSE_OFF + i];
    red[tid] = s;
    __syncthreads();
    for (int k = 128; k > 0; k >>= 1) {
        if (tid < k) red[tid] += red[tid + k];
        __syncthreads();
    }
    if (tid == 0) mse = red[0] / (float)MSE_N;
    __syncthreads();

    s = 0.0f;
    for (int i = tid; i < ATT_ROWS; i += 256) s += ws[REG_OFF + i];
    red[tid] = s;
    __syncthreads();
    for (int k = 128; k > 0; k >>= 1) {
        if (tid < k) red[tid] += red[tid + k];
        __syncthreads();
    }
    if (tid == 0) reg = red[0];
    __syncthreads();

    s = 0.0f;
    for (int i = tid; i < NTILE_PAIRS; i += 256) s += ws[COS_OFF + i];
    red[tid] = s;
    __syncthreads();
    for (int k = 128; k > 0; k >>= 1) {
        if (tid < k) red[tid] += red[tid + k];
        __syncthreads();
    }
    if (tid == 0) {
        cosv = red[0];
        out[0] = mse - REG_PARAM * reg + cosv;
    }
}

// ---------------------------------------------------------------------------
extern "C" void kernel_launch(void* const* d_in, const int* in_sizes, int n_in,
                              void* d_out, int out_size, void* d_ws, size_t ws_size,
                              hipStream_t stream) {
    const float* output = (const float*)d_in[0];   // (32,3,256,256)
    const float* gtruth = (const float*)d_in[1];   // (32,3,256,256)
    const float* att    = (const float*)d_in[2];   // (8192,2000)
    const float* mem    = (const float*)d_in[3];   // (2000,256)
    float* ws_f   = (float*)d_ws;
    float* result = (float*)d_out;

    mse_kernel    <<<MSE_BLOCKS, 256, 0, stream>>>(output, gtruth, ws_f + MSE_OFF);
    entropy_kernel<<<ATT_ROWS,   256, 0, stream>>>(att, ws_f + REG_OFF);
    invnorm_kernel<<<N_MEM,      256, 0, stream>>>(mem, ws_f + INV_OFF);
    gram_kernel   <<<NTILE_PAIRS, 32, 0, stream>>>(mem, ws_f + INV_OFF, ws_f + COS_OFF);
    final_kernel  <<<1,          256, 0, stream>>>(ws_f, result);
}